// UniAnchorGNN_48026324304370
// MI455X (gfx1250) — compile-verified
//
#include <hip/hip_runtime.h>
#include <hip/hip_bf16.h>

typedef float v2f __attribute__((ext_vector_type(2)));
typedef float v8f __attribute__((ext_vector_type(8)));

#define NN 50000
#define EE 800000
#define NGR 128
#define DD 128
#define MM 2
#define NTK 10

// ---------------- threefry2x32 (JAX-compatible) ----------------
__device__ __forceinline__ void tf_round(unsigned &x0, unsigned &x1, int r) {
    x0 += x1;
    x1 = (x1 << r) | (x1 >> (32 - r));
    x1 ^= x0;
}
__device__ __forceinline__ void threefry2x32(unsigned k0, unsigned k1,
                                             unsigned x0, unsigned x1,
                                             unsigned &o0, unsigned &o1) {
    unsigned k2 = k0 ^ k1 ^ 0x1BD11BDAu;
    x0 += k0; x1 += k1;
    tf_round(x0,x1,13); tf_round(x0,x1,15); tf_round(x0,x1,26); tf_round(x0,x1,6);
    x0 += k1; x1 += k2 + 1u;
    tf_round(x0,x1,17); tf_round(x0,x1,29); tf_round(x0,x1,16); tf_round(x0,x1,24);
    x0 += k2; x1 += k0 + 2u;
    tf_round(x0,x1,13); tf_round(x0,x1,15); tf_round(x0,x1,26); tf_round(x0,x1,6);
    x0 += k0; x1 += k1 + 3u;
    tf_round(x0,x1,17); tf_round(x0,x1,29); tf_round(x0,x1,16); tf_round(x0,x1,24);
    x0 += k1; x1 += k2 + 4u;
    tf_round(x0,x1,13); tf_round(x0,x1,15); tf_round(x0,x1,26); tf_round(x0,x1,6);
    x0 += k2; x1 += k0 + 5u;
    o0 = x0; o1 = x1;
}

// ---------------- embedding gather: tx[n] = x_table[x_idx[n]] ----------------
__global__ void k_embed(const int* __restrict__ xidx, const float* __restrict__ xt,
                        float* __restrict__ tx, int n_nodes) {
    int t = blockIdx.x * blockDim.x + threadIdx.x;     // n_nodes*32 threads (float4)
    if (t >= n_nodes * 32) return;
    int n = t >> 5, c4 = (t & 31) * 4;
    *(float4*)(tx + (size_t)n * DD + c4) =
        *(const float4*)(xt + (size_t)xidx[n] * DD + c4);
}

// ---------------- edge scatter: agg[dst] += h[src] (atomic f32) ----------------
__global__ void k_scatter(const float* __restrict__ hsrc, float* agg,
                          const int* __restrict__ esrc, const int* __restrict__ edst,
                          int n_edges) {
    int t = blockIdx.x * blockDim.x + threadIdx.x;     // n_edges*32 threads
    if (t >= n_edges * 32) return;
    int e = t >> 5, dg = (t & 31) * 4;
    int s = esrc[e], d = edst[e];
    float4 v = *(const float4*)(hsrc + (size_t)s * DD + dg);
    float* p = agg + (size_t)d * DD + dg;
    atomicAdd(p + 0, v.x); atomicAdd(p + 1, v.y);
    atomicAdd(p + 2, v.z); atomicAdd(p + 3, v.w);
}

// ---------------- WMMA GEMM: out = relu((in[+agg]) @ W + bias), cols=K=128 ----
// Block: 256 threads (8 waves), 64 rows. Wave w -> 16-col strip. In-place safe.
__global__ __launch_bounds__(256) void k_gemm128(const float* in, const float* agg,
                                                 const float* __restrict__ W,
                                                 const float* __restrict__ bias,
                                                 float* out, int R) {
    __shared__ float Al[64 * 130];                     // +2 pad: 8B-aligned v2f, no conflicts
    int tid = threadIdx.x;
    int rowBase = blockIdx.x * 64;

    // stage A tile (input + optional agg) into LDS, float4 vectorized
    for (int i = tid; i < 64 * 32; i += 256) {
        int r = i >> 5;
        int c4 = (i & 31) * 4;
        int row = rowBase + r;
        float4 v = make_float4(0.f, 0.f, 0.f, 0.f);
        if (row < R) {
            v = *(const float4*)(in + (size_t)row * DD + c4);
            if (agg) {
                float4 w2 = *(const float4*)(agg + (size_t)row * DD + c4);
                v.x += w2.x; v.y += w2.y; v.z += w2.z; v.w += w2.w;
            }
        }
        float* dst = &Al[r * 130 + c4];
        dst[0] = v.x; dst[1] = v.y; dst[2] = v.z; dst[3] = v.w;
    }
    __syncthreads();

    int wave = tid >> 5;
    int lane = tid & 31;
    int halfSel = lane >> 4;                           // k-pair select per ISA f32 layout
    int l16 = lane & 15;
    int colBase = wave * 16;

    // register-cache this wave's 128x16 B strip (64 VGPRs/lane)
    v2f breg[32];
#pragma unroll
    for (int kk = 0; kk < 32; ++kk) {
        int kp = kk * 4 + 2 * halfSel;
        breg[kk].x = W[(size_t)kp * DD + colBase + l16];
        breg[kk].y = W[(size_t)(kp + 1) * DD + colBase + l16];
    }
    float bv = bias[colBase + l16];

    for (int rt = 0; rt < 4; ++rt) {
        v8f c = {};
        const float* Arow = &Al[(rt * 16 + l16) * 130];
#pragma unroll
        for (int kk = 0; kk < 32; ++kk) {
            int kp = kk * 4 + 2 * halfSel;
            v2f a = *(const v2f*)(Arow + kp);          // ds_load_b64
            c = __builtin_amdgcn_wmma_f32_16x16x4_f32(
                false, a, false, breg[kk], (short)0, c, false, false);
        }
        int row0 = rowBase + rt * 16 + 8 * halfSel;
        int col = colBase + l16;
#pragma unroll
        for (int j = 0; j < 8; ++j) {
            int row = row0 + j;
            if (row < R) out[(size_t)row * DD + col] = fmaxf(c[j] + bv, 0.0f);
        }
    }
}

// ---------------- pred[n] = h[n] . Wd + bd (one wave per node) ----------------
__global__ void k_pred(const float* __restrict__ h, const float* __restrict__ Wd,
                       const float* __restrict__ bd, float* __restrict__ pred,
                       int n_nodes) {
    int w = (blockIdx.x * blockDim.x + threadIdx.x) >> 5;
    int lane = threadIdx.x & 31;
    if (w >= n_nodes) return;
    const float* row = h + (size_t)w * DD;
    float s = 0.f;
    for (int d = lane; d < DD; d += 32) s += row[d] * Wd[d];
    for (int off = 16; off; off >>= 1) s += __shfl_xor(s, off, 32);
    if (lane == 0) pred[w] = s + bd[0];
}

// ---------------- per-graph segment starts (batch is sorted) ----------------
__global__ void k_gstart(const int* __restrict__ batch, int* __restrict__ gstart,
                         int n_nodes, int ng) {
    int g = blockIdx.x * blockDim.x + threadIdx.x;
    if (g > ng) return;
    if (g == ng) { gstart[g] = n_nodes; return; }
    int lo = 0, hi = n_nodes;
    while (lo < hi) { int mid = (lo + hi) >> 1; if (batch[mid] < g) lo = mid + 1; else hi = mid; }
    gstart[g] = lo;
}

// ------------- per-graph softmax + Gumbel-max anchor selection --------------
__global__ __launch_bounds__(256) void k_anchor(const float* __restrict__ pred,
                                                const int* __restrict__ gstart,
                                                int* __restrict__ rs,
                                                int n_nodes, int half) {
    __shared__ float red[256];
    __shared__ int redi[256];
    int g = blockIdx.x, tid = threadIdx.x;
    int gs = gstart[g], ge = gstart[g + 1];

    float mx = -INFINITY;
    for (int n = gs + tid; n < ge; n += 256) mx = fmaxf(mx, pred[n]);
    red[tid] = mx; __syncthreads();
    for (int s2 = 128; s2; s2 >>= 1) { if (tid < s2) red[tid] = fmaxf(red[tid], red[tid + s2]); __syncthreads(); }
    mx = red[0]; __syncthreads();

    float ss = 0.f;
    for (int n = gs + tid; n < ge; n += 256) ss += expf(pred[n] - mx);
    red[tid] = ss; __syncthreads();
    for (int s2 = 128; s2; s2 >>= 1) { if (tid < s2) red[tid] += red[tid + s2]; __syncthreads(); }
    ss = red[0]; __syncthreads();

    unsigned fk0, fk1;                                 // fold_in(key(1), 1)
    threefry2x32(0u, 1u, 0u, 1u, fk0, fk1);

    for (int m = 0; m < MM; ++m) {
        float bs = -INFINITY; int bi = n_nodes;
        for (int n = gs + tid; n < ge; n += 256) {
            float lp = logf(expf(pred[n] - mx) + 1e-15f * ss);  // log(prob+1e-15) + const
            unsigned i = (unsigned)(m * n_nodes + n);
            unsigned o0, o1, bits;
            if (i < (unsigned)half) { threefry2x32(fk0, fk1, i, i + half, o0, o1); bits = o0; }
            else                    { threefry2x32(fk0, fk1, i - half, i, o0, o1); bits = o1; }
            float u = __uint_as_float(0x3F800000u | (bits >> 9)) - 1.0f;
            float gum = -logf(-logf(u + 1e-12f) + 1e-12f);
            float sc = lp + gum;
            if (sc > bs || (sc == bs && n < bi)) { bs = sc; bi = n; }
        }
        red[tid] = bs; redi[tid] = bi; __syncthreads();
        for (int s2 = 128; s2; s2 >>= 1) {
            if (tid < s2) {
                if (red[tid + s2] > red[tid] ||
                    (red[tid + s2] == red[tid] && redi[tid + s2] < redi[tid])) {
                    red[tid] = red[tid + s2]; redi[tid] = redi[tid + s2];
                }
            }
            __syncthreads();
        }
        if (tid == 0) rs[m * NGR + g] = (ge > gs) ? redi[0] : n_nodes;
        __syncthreads();
    }
}

// -------- build x per copy: h[m,n] = tx[n] * (1 + A1 * is_anchor) -----------
__global__ void k_buildx(const float* __restrict__ tx, const float* __restrict__ at,
                         const int* __restrict__ batch, const int* __restrict__ rs,
                         float* __restrict__ h, int n_nodes) {
    int t = blockIdx.x * blockDim.x + threadIdx.x;     // M*n_nodes*32 threads
    if (t >= MM * n_nodes * 32) return;
    int m = t / (n_nodes * 32);
    int rem = t - m * n_nodes * 32;
    int n = rem >> 5, c4 = (rem & 31) * 4;
    float4 v = *(const float4*)(tx + (size_t)n * DD + c4);
    if (rs[m * NGR + batch[n]] == n) {
        float4 a1 = *(const float4*)(at + DD + c4);    // anchor_table row 1
        v.x *= (1.f + a1.x); v.y *= (1.f + a1.y);
        v.z *= (1.f + a1.z); v.w *= (1.f + a1.w);
    }
    *(float4*)(h + ((size_t)m * n_nodes + n) * DD + c4) = v;
}

// ---------------- per-graph mean pool: hg[m,g] = mean_n hn[m,n] -------------
__global__ void k_pool(const float* __restrict__ hn, const int* __restrict__ gstart,
                       float* __restrict__ hg, int n_nodes) {
    int b = blockIdx.x;                                // m*NGR + g
    int m = b / NGR, g = b % NGR;
    int d = threadIdx.x;                               // 128 threads
    int gs = gstart[g], ge = gstart[g + 1];
    float s = 0.f;
    for (int n = gs; n < ge; ++n) s += hn[((size_t)m * n_nodes + n) * DD + d];
    hg[((size_t)m * NGR + g) * DD + d] = s / fmaxf((float)(ge - gs), 1.0f);
}

// ---------------- final: out[g,t] = bp[t] + mean_m hg[m,g].Wp[:,t] ----------
__global__ void k_final(const float* __restrict__ hg, const float* __restrict__ Wp,
                        const float* __restrict__ bp, float* __restrict__ out) {
    int w = (blockIdx.x * blockDim.x + threadIdx.x) >> 5;
    int lane = threadIdx.x & 31;
    if (w >= NGR * NTK) return;
    int g = w / NTK, t = w % NTK;
    float s = 0.f;
    for (int m = 0; m < MM; ++m)
        for (int d = lane; d < DD; d += 32)
            s += hg[((size_t)m * NGR + g) * DD + d] * Wp[d * NTK + t];
    for (int off = 16; off; off >>= 1) s += __shfl_xor(s, off, 32);
    if (lane == 0) out[g * NTK + t] = bp[t] + s * (1.0f / MM);
}

extern "C" void kernel_launch(void* const* d_in, const int* in_sizes, int n_in,
                              void* d_out, int out_size, void* d_ws, size_t ws_size,
                              hipStream_t stream) {
    const int*   x_idx        = (const int*)d_in[0];
    const int*   esrc         = (const int*)d_in[1];
    const int*   edst         = (const int*)d_in[2];
    const int*   batch        = (const int*)d_in[3];
    const float* x_table      = (const float*)d_in[4];
    const float* anchor_table = (const float*)d_in[5];
    const float* Wg           = (const float*)d_in[6];
    const float* bg           = (const float*)d_in[7];
    const float* Wn           = (const float*)d_in[8];
    const float* bn           = (const float*)d_in[9];
    const float* Wd           = (const float*)d_in[10];
    const float* bd           = (const float*)d_in[11];
    const float* Wp           = (const float*)d_in[12];
    const float* bp           = (const float*)d_in[13];
    float* out = (float*)d_out;

    const int N = NN, E = EE, D = DD;
    float* ws   = (float*)d_ws;
    float* tx   = ws;                                  // N*D
    float* h    = tx + (size_t)N * D;                  // M*N*D
    float* agg  = h + (size_t)MM * N * D;              // M*N*D
    float* pred = agg + (size_t)MM * N * D;            // N
    int*   gstart = (int*)(pred + N);                  // NGR+1
    int*   rs   = gstart + (NGR + 4);                  // M*NGR
    float* hg   = (float*)(rs + MM * NGR);             // M*NGR*D

    // embedding + graph segment boundaries
    k_embed<<<(N * 32 + 255) / 256, 256, 0, stream>>>(x_idx, x_table, tx, N);
    k_gstart<<<1, 256, 0, stream>>>(batch, gstart, N, NGR);

    // ---- GNN pass 1 (M copies identical -> compute once) ----
    hipMemsetAsync(agg, 0, (size_t)N * D * sizeof(float), stream);
    k_scatter<<<(E * 32 + 255) / 256, 256, 0, stream>>>(tx, agg, esrc, edst, E);
    k_gemm128<<<(N + 63) / 64, 256, 0, stream>>>(tx, agg, Wg, bg, h, N);

    hipMemsetAsync(agg, 0, (size_t)N * D * sizeof(float), stream);
    k_scatter<<<(E * 32 + 255) / 256, 256, 0, stream>>>(h, agg, esrc, edst, E);
    k_gemm128<<<(N + 63) / 64, 256, 0, stream>>>(h, agg, Wg + D * D, bg + D, h, N);

    // anchor sampling (per-graph softmax + Gumbel-max via threefry)
    k_pred<<<(N * 32 + 255) / 256, 256, 0, stream>>>(h, Wd, bd, pred, N);
    k_anchor<<<NGR, 256, 0, stream>>>(pred, gstart, rs, N, MM * N / 2);

    // x = tx * at_eff[anchor] + tx, per copy
    k_buildx<<<(MM * N * 32 + 255) / 256, 256, 0, stream>>>(tx, anchor_table, batch, rs, h, N);

    // ---- GNN pass 2 (M=2 copies) ----
    for (int l = 0; l < 2; ++l) {
        hipMemsetAsync(agg, 0, (size_t)MM * N * D * sizeof(float), stream);
        for (int m = 0; m < MM; ++m)
            k_scatter<<<(E * 32 + 255) / 256, 256, 0, stream>>>(
                h + (size_t)m * N * D, agg + (size_t)m * N * D, esrc, edst, E);
        k_gemm128<<<(MM * N + 63) / 64, 256, 0, stream>>>(
            h, agg, Wg + (size_t)l * D * D, bg + (size_t)l * D, h, MM * N);
    }

    // node MLP (no agg), in-place
    k_gemm128<<<(MM * N + 63) / 64, 256, 0, stream>>>(h, nullptr, Wn, bn, h, MM * N);

    // pool + readout
    k_pool<<<MM * NGR, 128, 0, stream>>>(h, gstart, hg, N);
    k_final<<<(NGR * NTK * 32 + 255) / 256, 256, 0, stream>>>(hg, Wp, bp, out);
}